// QuantumEmbedding_27161373180356
// MI455X (gfx1250) — compile-verified
//
#include <hip/hip_runtime.h>

typedef float v2f __attribute__((ext_vector_type(2)));
typedef float v4f __attribute__((ext_vector_type(4)));
typedef float v8f __attribute__((ext_vector_type(8)));

#define BQ 8
#define SQ 4096
#define DQ 768
#define NQ 10
#define ROWS (BQ*SQ)            // 32768
#define NSTATE 1024
#define WPAD 772                // padded LDS row stride (bank-conflict-free b64 reads)
#define WAVES 8
#define ROWS_PER_WAVE 16
#define ROWS_PER_BLOCK (WAVES*ROWS_PER_WAVE)   // 128

__global__ __launch_bounds__(256)
void qembed_kernel(const float* __restrict__ vt,
                   const float* __restrict__ Wpr,
                   float* __restrict__ out)
{
  __shared__ float lds_w[16*WPAD];      // Wpr padded to 16 rows x 772
  __shared__ float lds_ang[WAVES*256];  // per-wave 16x16 angle tile

  const int tid  = threadIdx.x;
  const int wave = tid >> 5;
  const int lane = tid & 31;
  const int n    = lane & 15;   // N column / A row within tile
  const int g    = lane >> 4;   // K half selector (0: K=0,1  1: K=2,3)

  // ---- stage Wpr into LDS (rows 10..15 zero-padded) ----
  for (int i = tid; i < 16*WPAD; i += 256) lds_w[i] = 0.0f;
  __syncthreads();
  for (int i = tid; i < NQ*DQ; i += 256) {
    int q = i / DQ;
    int d = i - q*DQ;
    lds_w[q*WPAD + d] = Wpr[i];
  }
  __syncthreads();

  const int rowTile = blockIdx.x * ROWS_PER_BLOCK + wave * ROWS_PER_WAVE;

  // ---- phase 1: angles = vt * Wpr^T via V_WMMA_F32_16X16X4_F32 ----
  // A 16x4 f32 layout: lanes 0-15 hold M=lane, VGPR0=K0,VGPR1=K1;
  //                    lanes 16-31 hold M=lane-16, VGPR0=K2,VGPR1=K3.
  const float* aRow = vt + (size_t)(rowTile + n) * DQ + 2*g;
  const float* bRow = &lds_w[n*WPAD + 2*g];

  v8f acc0 = {0.f,0.f,0.f,0.f,0.f,0.f,0.f,0.f};
  v8f acc1 = {0.f,0.f,0.f,0.f,0.f,0.f,0.f,0.f};
  #pragma unroll 8
  for (int k0 = 0; k0 < DQ; k0 += 8) {
    v2f a0 = *(const v2f*)(aRow + k0);
    v2f b0 = *(const v2f*)(bRow + k0);
    acc0 = __builtin_amdgcn_wmma_f32_16x16x4_f32(false, a0, false, b0,
                                                 (short)0, acc0, false, false);
    v2f a1 = *(const v2f*)(aRow + k0 + 4);
    v2f b1 = *(const v2f*)(bRow + k0 + 4);
    acc1 = __builtin_amdgcn_wmma_f32_16x16x4_f32(false, a1, false, b1,
                                                 (short)0, acc1, false, false);
  }
  v8f acc = acc0 + acc1;

  // scatter accumulator (angles, 16x16 row-major) into per-wave LDS scratch.
  // D layout: VGPR r -> M = r + 8*g, N = lane&15.
  {
    int base = wave*256 + (8*g)*16 + n;
    #pragma unroll
    for (int r = 0; r < 8; ++r)
      lds_ang[base + r*16] = acc[r];
  }
  __syncthreads();

  // ---- phase 2: build product state, 1 row at a time, whole wave per row ----
  // Output index idx (bit9..bit0 = wire0..wire9) decomposed as:
  //   idx = i5*128 + lane*4 + t
  //   i5 (bits 9:7)  -> wires 0,1,2   (h8 table, 8 values)
  //   lane bits 4:3  -> wires 3,4     (hbase, per-lane constant)
  //   lane bits 2:0  -> wires 5,6,7   (lbase, per-lane constant)
  //   t (bits 1:0)   -> wires 8,9     (vlo0..3)
  const int g2   = lane >> 3;
  const int t567 = lane & 7;

  for (int m = 0; m < ROWS_PER_WAVE; ++m) {
    const float* ap = &lds_ang[wave*256 + m*16];   // broadcast LDS reads
    float c[NQ], s[NQ];
    #pragma unroll
    for (int q = 0; q < NQ; ++q) {
      float h = 0.5f * ap[q];
      c[q] = __cosf(h);
      s[q] = __sinf(h);
    }
    float hbase = ((g2 & 2) ? s[3] : c[3]) * ((g2 & 1) ? s[4] : c[4]);
    float lbase = ((t567 & 4) ? s[5] : c[5]) * ((t567 & 2) ? s[6] : c[6])
                * ((t567 & 1) ? s[7] : c[7]);
    float vlo0 = lbase * (c[8]*c[9]);
    float vlo1 = lbase * (c[8]*s[9]);
    float vlo2 = lbase * (s[8]*c[9]);
    float vlo3 = lbase * (s[8]*s[9]);

    float a0 = c[0]*hbase, a1 = s[0]*hbase;
    float b00 = a0*c[1], b01 = a0*s[1], b10 = a1*c[1], b11 = a1*s[1];
    float h8[8];
    h8[0]=b00*c[2]; h8[1]=b00*s[2];
    h8[2]=b01*c[2]; h8[3]=b01*s[2];
    h8[4]=b10*c[2]; h8[5]=b10*s[2];
    h8[6]=b11*c[2]; h8[7]=b11*s[2];

    float* rowout = out + (size_t)(rowTile + m) * NSTATE + lane*4;
    #pragma unroll
    for (int i5 = 0; i5 < 8; ++i5) {
      float hv = h8[i5];
      v4f o = {hv*vlo0, hv*vlo1, hv*vlo2, hv*vlo3};
      *(v4f*)(rowout + i5*128) = o;     // coalesced global_store_b128
    }
  }

  // ---- write angles output [ROWS, NQ] (second tuple element) ----
  float* angOut = out + (size_t)ROWS * NSTATE;
  #pragma unroll
  for (int i = 0; i < 5; ++i) {
    int j = lane*5 + i;          // 0..159 over 16 rows x 10 angles
    int r = j / NQ;
    int q = j - r*NQ;
    angOut[(size_t)(rowTile + r)*NQ + q] = lds_ang[wave*256 + r*16 + q];
  }
}

extern "C" void kernel_launch(void* const* d_in, const int* in_sizes, int n_in,
                              void* d_out, int out_size, void* d_ws, size_t ws_size,
                              hipStream_t stream) {
  const float* vt  = (const float*)d_in[0];
  const float* Wpr = (const float*)d_in[1];
  float* out = (float*)d_out;
  dim3 grid(ROWS / ROWS_PER_BLOCK);   // 256 blocks
  dim3 block(256);                    // 8 wave32 per block
  qembed_kernel<<<grid, block, 0, stream>>>(vt, Wpr, out);
}